// KMeans_5102421147695
// MI455X (gfx1250) — compile-verified
//
#include <hip/hip_runtime.h>

typedef __attribute__((ext_vector_type(2))) float v2f;
typedef __attribute__((ext_vector_type(8))) float v8f;

#define LDA 132            // padded LDS row stride (floats): breaks stride-128 bank conflicts
#define B_DIM 4
#define H_DIM 16
#define L_DIM 4096
#define D_DIM 128
#define C_DIM 512
#define COMMIT_SCALE (1.0e-4f / 33554432.0f)   // COMMITMENT / (b*h*l*d)

__global__ void kmeans_zero_loss(float* loss) {
    if (threadIdx.x == 0) loss[0] = 0.0f;
}

__global__ __launch_bounds__(128)
void kmeans_route_kernel(const float* __restrict__ x,
                         const float* __restrict__ means,
                         float* __restrict__ dists,
                         float* __restrict__ loss)
{
    __shared__ float As[16 * LDA];   // normalized x tile, 16 rows x 128 (padded)
    __shared__ float Bs[64 * LDA];   // means chunk, 64 rows x 128 (padded)
    __shared__ float redV[64];
    __shared__ int   redI[64];
    __shared__ int   bucketS[16];

    const int t    = threadIdx.x;
    const int wave = t >> 5;
    const int lane = t & 31;
    const int hi   = lane >> 4;      // half-wave select (K/M-row split per WMMA layout)
    const int lm   = lane & 15;

    const int bh       = blockIdx.y;          // b*16 + h
    const int h        = bh & (H_DIM - 1);
    const int row_base = blockIdx.x * 16;

    const float* xblk    = x + ((size_t)bh * L_DIM + row_base) * D_DIM;
    const float* means_h = means + (size_t)h * C_DIM * D_DIM;
    float*       dout    = dists + ((size_t)bh * L_DIM + row_base) * C_DIM;

    // ---- stage x tile: 16 rows x 128 floats, 8 threads per row, float4 loads ----
    {
        const int r  = t >> 3;
        const int c0 = (t & 7) * 16;
        const float4* src = (const float4*)(xblk + (size_t)r * D_DIM + c0);
        float4* dst = (float4*)&As[r * LDA + c0];
        #pragma unroll
        for (int i = 0; i < 4; ++i) dst[i] = src[i];
    }
    __syncthreads();

    // ---- L2-normalize rows in LDS (each wave owns 4 rows; wave32 shuffle reduce) ----
    #pragma unroll
    for (int rr = 0; rr < 4; ++rr) {
        const int m = wave * 4 + rr;
        float4 v = *(float4*)&As[m * LDA + lane * 4];
        float ss = v.x * v.x + v.y * v.y + v.z * v.z + v.w * v.w;
        #pragma unroll
        for (int mk = 16; mk >= 1; mk >>= 1) ss += __shfl_xor(ss, mk, 32);
        const float scale = 1.0f / fmaxf(sqrtf(ss), 1e-12f);
        v.x *= scale; v.y *= scale; v.z *= scale; v.w *= scale;
        *(float4*)&As[m * LDA + lane * 4] = v;
    }
    __syncthreads();

    float best[8];
    int   bidx[8];
    #pragma unroll
    for (int r = 0; r < 8; ++r) { best[r] = -3.402823466e38f; bidx[r] = 0x7fffffff; }

    const int nl = wave * 16 + lm;   // this lane's local column within a 64-col chunk

    #pragma unroll 1
    for (int ch = 0; ch < 8; ++ch) {
        const int n0 = ch * 64;
        // ---- stage means chunk: 64 rows x 128 floats, 2 threads per row ----
        {
            const int br = t >> 1;
            const int bc = (t & 1) * 64;
            const float4* src = (const float4*)(means_h + (size_t)(n0 + br) * D_DIM + bc);
            float4* dst = (float4*)&Bs[br * LDA + bc];
            #pragma unroll
            for (int i = 0; i < 16; ++i) dst[i] = src[i];
        }
        __syncthreads();

        // ---- 16x16 f32 tile per wave, K=128 via 32x v_wmma_f32_16x16x4_f32 ----
        v8f cacc = {0.f, 0.f, 0.f, 0.f, 0.f, 0.f, 0.f, 0.f};
        #pragma unroll
        for (int kt = 0; kt < 32; ++kt) {
            const int kk = kt * 4 + hi * 2;                    // A/B K-pair per ISA layout
            v2f av = *(const v2f*)&As[lm * LDA + kk];          // A: lane = M row
            v2f bv = *(const v2f*)&Bs[nl * LDA + kk];          // B: lane = N col
            cacc = __builtin_amdgcn_wmma_f32_16x16x4_f32(
                false, av, false, bv, (short)0, cacc, false, false);
        }

        // ---- fused argmax update + dists store (VGPR r = rows r / r+8) ----
        const int colg = n0 + nl;
        #pragma unroll
        for (int r = 0; r < 8; ++r) {
            const float v = cacc[r];
            if (v > best[r] || (v == best[r] && colg < bidx[r])) {
                best[r] = v; bidx[r] = colg;
            }
            dout[(size_t)(r + hi * 8) * C_DIM + colg] = v;
        }
        __syncthreads();   // Bs is reloaded next iteration
    }

    // ---- argmax reduction within 16-lane halves (rows live per-half) ----
    #pragma unroll
    for (int r = 0; r < 8; ++r) {
        float bv = best[r]; int bi = bidx[r];
        #pragma unroll
        for (int mk = 1; mk <= 8; mk <<= 1) {
            const float ov = __shfl_xor(bv, mk, 32);
            const int   oi = __shfl_xor(bi, mk, 32);
            if (ov > bv || (ov == bv && oi < bi)) { bv = ov; bi = oi; }
        }
        if (lm == 0) {
            const int row = r + hi * 8;
            redV[row * 4 + wave] = bv;
            redI[row * 4 + wave] = bi;
        }
    }
    __syncthreads();

    // ---- merge the 4 waves' candidates per row ----
    if (t < 16) {
        float bv = redV[t * 4]; int bi = redI[t * 4];
        #pragma unroll
        for (int w = 1; w < 4; ++w) {
            const float ov = redV[t * 4 + w];
            const int   oi = redI[t * 4 + w];
            if (ov > bv || (ov == bv && oi < bi)) { bv = ov; bi = oi; }
        }
        bucketS[t] = bi;
    }
    __syncthreads();

    // ---- fused commitment loss: sum (xn - means[h, bucket])^2 over tile ----
    float acc = 0.f;
    #pragma unroll
    for (int rr = 0; rr < 4; ++rr) {
        const int m  = wave * 4 + rr;
        const int bk = bucketS[m];
        const float4 xv = *(float4*)&As[m * LDA + lane * 4];
        const float4 mv = ((const float4*)(means_h + (size_t)bk * D_DIM))[lane];
        const float dx = xv.x - mv.x, dy = xv.y - mv.y;
        const float dz = xv.z - mv.z, dw = xv.w - mv.w;
        acc += dx * dx + dy * dy + dz * dz + dw * dw;
    }
    #pragma unroll
    for (int mk = 16; mk >= 1; mk >>= 1) acc += __shfl_xor(acc, mk, 32);
    if (lane == 0) atomicAdd(loss, acc * COMMIT_SCALE);
}

extern "C" void kernel_launch(void* const* d_in, const int* in_sizes, int n_in,
                              void* d_out, int out_size, void* d_ws, size_t ws_size,
                              hipStream_t stream) {
    const float* x     = (const float*)d_in[0];   // [4,16,4096,128] f32
    const float* means = (const float*)d_in[1];   // [16,512,128]   f32
    float* dists = (float*)d_out;                               // [4,16,4096,512]
    float* loss  = dists + (size_t)B_DIM * H_DIM * L_DIM * C_DIM;  // scalar after dists

    kmeans_zero_loss<<<dim3(1), dim3(32), 0, stream>>>(loss);

    dim3 grid(L_DIM / 16, B_DIM * H_DIM);
    kmeans_route_kernel<<<grid, dim3(128), 0, stream>>>(x, means, dists, loss);
}